// WriteHead_77240691851638
// MI455X (gfx1250) — compile-verified
//
#include <hip/hip_runtime.h>
#include <math.h>

#define HH 4
#define WW 64
#define NN 8192
#define NV (NN / 4)                      // float4 elements per head row
#define EPSF 1e-8f
#define CTRL_PER_B (HH * (3 * WW + 6))   // 792

typedef float v2f __attribute__((ext_vector_type(2)));
typedef float v8f __attribute__((ext_vector_type(8)));

__device__ __forceinline__ float softplusf_(float x) {
    return (x > 20.f) ? x : log1pf(expf(x));
}
__device__ __forceinline__ float sigmoidf_(float x) {
    return 1.f / (1.f + expf(-x));
}

// Block-wide reduction of 4 per-thread values (sum or max). red: 4*32 floats.
__device__ __forceinline__ void block_reduce4(const float v[4], float* red,
                                              float* outres, int t, int lane,
                                              int wid, bool ismax) {
    #pragma unroll
    for (int h = 0; h < 4; ++h) {
        float x = v[h];
        #pragma unroll
        for (int off = 16; off > 0; off >>= 1) {
            float o = __shfl_xor(x, off, 32);
            x = ismax ? fmaxf(x, o) : (x + o);
        }
        if (lane == 0) red[h * 32 + wid] = x;
    }
    __syncthreads();
    if (t < 4) {
        float a = ismax ? -INFINITY : 0.f;
        for (int i = 0; i < 32; ++i) {
            float x = red[t * 32 + i];
            a = ismax ? fmaxf(a, x) : (a + x);
        }
        outres[t] = a;
    }
    __syncthreads();
}

__global__ __launch_bounds__(1024)
void ntm_write_head_kernel(const float* __restrict__ memory,
                           const float* __restrict__ controls,
                           const float* __restrict__ prevw,
                           float* __restrict__ out) {
    extern __shared__ float sm[];
    const int b    = blockIdx.x;
    const int t    = threadIdx.x;
    const int lane = t & 31;
    const int wid  = t >> 5;
    const int HN   = HH * NN;

    float* A1    = sm;                 // [H][N] working weights (128 KB)
    float* tkp   = A1 + HN;            // tanh(keys) padded to 16 rows [16][W]
    float* ed    = tkp + 16 * WW;      // sigmoid(erase) [H][W]
    float* wrd   = ed + HH * WW;       // tanh(write)    [H][W]
    float* knorm = wrd + HH * WW;      // [H]
    float* sbeta = knorm + HH;         // [H]
    float* sgate = sbeta + HH;         // [H]
    float* sgam  = sgate + HH;         // [H]
    float* sshf  = sgam + HH;          // [H][3] softmaxed shifts
    float* smax  = sshf + 3 * HH;      // [H]
    float* sden  = smax + HH;          // [H] content-softmax sum
    float* sinv  = sden + HH;          // [H] 1/(sharp sum + eps)
    float* red   = sinv + HH;          // [4][32] reduction scratch
    float* dump  = red + 4 * 32;       // [4] dead-store target for inactive lanes

    const float* ctl   = controls + (size_t)b * CTRL_PER_B;
    const float* memb  = memory + (size_t)b * NN * WW;
    const float* prevb = prevw + (size_t)b * HN;
    float*       outb  = out + (size_t)b * NN * WW;

    // ---------- P0: control transforms ----------
    // tkp has 16 rows; rows 4..15 are zero so WMMA B columns 4..15 vanish.
    tkp[t] = (t < HH * WW) ? tanhf(ctl[t]) : 0.f;   // blockDim == 16*WW == 1024
    if (t < HH * WW) {
        ed[t]  = sigmoidf_(ctl[HH * WW + t]);
        wrd[t] = tanhf(ctl[2 * HH * WW + t]);
    } else if (t >= 512 && t < 512 + HH) {
        const int h    = t - 512;
        const int base = 3 * HH * WW;  // 768
        sbeta[h] = softplusf_(ctl[base + h]);
        sgate[h] = sigmoidf_(ctl[base + HH + h]);
        float s0 = ctl[base + 2 * HH + 3 * h + 0];
        float s1 = ctl[base + 2 * HH + 3 * h + 1];
        float s2 = ctl[base + 2 * HH + 3 * h + 2];
        float mx = fmaxf(s0, fmaxf(s1, s2));
        float e0 = expf(s0 - mx), e1 = expf(s1 - mx), e2 = expf(s2 - mx);
        float inv = 1.f / (e0 + e1 + e2);
        sshf[3 * h + 0] = e0 * inv;
        sshf[3 * h + 1] = e1 * inv;
        sshf[3 * h + 2] = e2 * inv;
        sgam[h] = 1.f + softplusf_(ctl[base + 5 * HH + h]);
    }
    __syncthreads();
    if (t < HH) {
        float s = 0.f;
        for (int w = 0; w < WW; ++w) { float v = tkp[t * WW + w]; s += v * v; }
        knorm[t] = sqrtf(s);
    }
    __syncthreads();

    // ---------- P1: scores via v_wmma_f32_16x16x4_f32 ----------
    // D[16 mem rows][16 cols (h in 0..3)] += A[16x4 mem chunk] * B[4x16 key chunk]
    {
        const int ntiles = NN >> 4;
        const int rowA   = lane & 15;         // A: M index
        const int khi    = (lane >> 4) << 1;  // A/B: K pair select (0 or 2)
        const int col    = lane & 15;         // B/D: N index
        const bool act   = col < HH;

        // Hoist the tile-invariant B fragments (keys) into registers.
        v2f bv[WW / 4];
        #pragma unroll
        for (int c = 0; c < WW / 4; ++c) {
            bv[c].x = tkp[col * WW + 4 * c + khi];
            bv[c].y = tkp[col * WW + 4 * c + khi + 1];
        }
        const float kn = knorm[col & 3];
        const float kb = sbeta[col & 3];

        for (int tile = wid; tile < ntiles; tile += 32) {
            const int n0 = tile << 4;
            const float* mrow = memb + (size_t)(n0 + rowA) * WW;
            v8f d = {};
            float rsq = 0.f;
            #pragma unroll
            for (int c = 0; c < WW / 4; ++c) {
                const int k0 = 4 * c + khi;
                v2f a;
                a.x = mrow[k0];
                a.y = mrow[k0 + 1];
                rsq += a.x * a.x + a.y * a.y;
                d = __builtin_amdgcn_wmma_f32_16x16x4_f32(
                        false, a, false, bv[c], (short)0, d, false, false);
            }
            float rfull = rsq + __shfl_xor(rsq, 16, 32);   // full |m_row|^2
            #pragma unroll
            for (int v = 0; v < 8; ++v) {
                const int r = v + ((lane >> 4) << 3);      // D row
                float msq = __shfl(rfull, r, 32);
                float sc  = kb * (d[v] / (sqrtf(msq) * kn + EPSF));
                // unconditional store: inactive lanes hit the dump word
                float* dst = act ? (A1 + col * NN + n0 + r) : dump;
                *dst = sc;
            }
        }
    }
    __syncthreads();

    // ---------- P2: per-head max over N (float4) ----------
    const float4* A1v4 = (const float4*)A1;
    float4*       A1w4 = (float4*)A1;
    float lv[4];
    #pragma unroll
    for (int h = 0; h < HH; ++h) {
        float m = -INFINITY;
        #pragma unroll
        for (int k = 0; k < NV / 1024; ++k) {
            float4 v = A1v4[h * NV + t + k * 1024];
            m = fmaxf(m, fmaxf(fmaxf(v.x, v.y), fmaxf(v.z, v.w)));
        }
        lv[h] = m;
    }
    block_reduce4(lv, red, smax, t, lane, wid, true);

    // ---------- P3: exp + sum (content softmax numerator, in place) ----------
    #pragma unroll
    for (int h = 0; h < HH; ++h) {
        const float mx = smax[h];
        float s = 0.f;
        #pragma unroll
        for (int k = 0; k < NV / 1024; ++k) {
            const int i = h * NV + t + k * 1024;
            float4 v = A1v4[i];
            v.x = expf(v.x - mx); v.y = expf(v.y - mx);
            v.z = expf(v.z - mx); v.w = expf(v.w - mx);
            A1w4[i] = v;
            s += v.x + v.y + v.z + v.w;
        }
        lv[h] = s;
    }
    block_reduce4(lv, red, sden, t, lane, wid, false);

    // ---------- P4: normalize + gate-interpolate with prev_weights ----------
    const float4* prev4 = (const float4*)prevb;
    #pragma unroll
    for (int h = 0; h < HH; ++h) {
        const float g    = sgate[h];
        const float gi   = g / sden[h];
        const float omg  = 1.f - g;
        #pragma unroll
        for (int k = 0; k < NV / 1024; ++k) {
            const int i = h * NV + t + k * 1024;
            float4 v = A1v4[i];
            float4 p = prev4[i];
            v.x = gi * v.x + omg * p.x;
            v.y = gi * v.y + omg * p.y;
            v.z = gi * v.z + omg * p.z;
            v.w = gi * v.w + omg * p.w;
            A1w4[i] = v;
        }
    }
    __syncthreads();

    // ---------- P5: circular 3-tap shift + sharpen (per head, in place) ----------
    #pragma unroll
    for (int h = 0; h < HH; ++h) {
        const float s0 = sshf[3 * h + 0];
        const float s1 = sshf[3 * h + 1];
        const float s2 = sshf[3 * h + 2];
        const float gm = sgam[h];
        float4 tmp[NV / 1024];
        float acc = 0.f;
        #pragma unroll
        for (int k = 0; k < NV / 1024; ++k) {
            const int i4 = t + k * 1024;
            const int nb = i4 << 2;
            float4 v = A1v4[h * NV + i4];
            float wm = A1[h * NN + ((nb + NN - 1) & (NN - 1))];
            float wp = A1[h * NN + ((nb + 4) & (NN - 1))];
            float4 o;
            o.x = s0 * wm  + s1 * v.x + s2 * v.y;
            o.y = s0 * v.x + s1 * v.y + s2 * v.z;
            o.z = s0 * v.y + s1 * v.z + s2 * v.w;
            o.w = s0 * v.z + s1 * v.w + s2 * wp;
            o.x = powf(o.x, gm); o.y = powf(o.y, gm);
            o.z = powf(o.z, gm); o.w = powf(o.w, gm);
            tmp[k] = o;
            acc += o.x + o.y + o.z + o.w;
        }
        __syncthreads();   // all reads of this head's row done
        #pragma unroll
        for (int k = 0; k < NV / 1024; ++k)
            A1w4[h * NV + t + k * 1024] = tmp[k];
        __syncthreads();
        lv[h] = acc;
    }
    block_reduce4(lv, red, sinv, t, lane, wid, false);
    if (t < HH) sinv[t] = 1.f / (sinv[t] + EPSF);
    __syncthreads();

    // ---------- P7: erase-product + update, stream memory -> out (b128) ----------
    {
        const int q  = t & 15;        // float4 column within row
        const int nr = t >> 4;        // row 0..63
        const float4* edv  = (const float4*)ed;
        const float4* wrv  = (const float4*)wrd;
        const float4* memv = (const float4*)memb;
        float4*       outv = (float4*)outb;
        float4 edr[4], wrr[4];
        float  iv[4];
        #pragma unroll
        for (int h = 0; h < HH; ++h) {
            edr[h] = edv[h * (WW / 4) + q];
            wrr[h] = wrv[h * (WW / 4) + q];
            iv[h]  = sinv[h];
        }
        for (int k = 0; k < NN / 64; ++k) {
            const int n = nr + (k << 6);
            const int i4 = n * (WW / 4) + q;
            float4 m = memv[i4];
            float ex = 1.f, ey = 1.f, ez = 1.f, ew = 1.f;
            float ux = 0.f, uy = 0.f, uz = 0.f, uw = 0.f;
            #pragma unroll
            for (int h = 0; h < HH; ++h) {
                float wdv = A1[h * NN + n] * iv[h];
                ex *= 1.f - wdv * edr[h].x;  ux += wdv * wrr[h].x;
                ey *= 1.f - wdv * edr[h].y;  uy += wdv * wrr[h].y;
                ez *= 1.f - wdv * edr[h].z;  uz += wdv * wrr[h].z;
                ew *= 1.f - wdv * edr[h].w;  uw += wdv * wrr[h].w;
            }
            float4 o;
            o.x = m.x * ex + ux;
            o.y = m.y * ey + uy;
            o.z = m.z * ez + uz;
            o.w = m.w * ew + uw;
            outv[i4] = o;
        }
    }
}

extern "C" void kernel_launch(void* const* d_in, const int* in_sizes, int n_in,
                              void* d_out, int out_size, void* d_ws, size_t ws_size,
                              hipStream_t stream) {
    const float* memory   = (const float*)d_in[0];
    const float* controls = (const float*)d_in[1];
    const float* prevw    = (const float*)d_in[2];
    float* out = (float*)d_out;

    const int B = in_sizes[1] / CTRL_PER_B;

    const size_t shmem =
        (size_t)(HH * NN + 16 * WW + 2 * HH * WW + 4 * HH + 3 * HH + 3 * HH +
                 4 * 32 + 4) * sizeof(float);

    // Opt in to >64KB dynamic LDS (CDNA5 WGP has 320KB). Idempotent, capture-safe.
    (void)hipFuncSetAttribute((const void*)ntm_write_head_kernel,
                              hipFuncAttributeMaxDynamicSharedMemorySize,
                              (int)shmem);

    ntm_write_head_kernel<<<dim3(B), dim3(1024), shmem, stream>>>(
        memory, controls, prevw, out);
}